// TorchQNC2Adapter_59416577573264
// MI455X (gfx1250) — compile-verified
//
#include <hip/hip_runtime.h>
#include <math.h>

// ---------------------------------------------------------------------------
// Analytic reduction of the 12-qubit PK/PD circuit:
//   out[b,w] = bounds(  x_b^T  M_w  x_b ),  x_b in R^16,  M_w 16x16 real.
// Kernel 1: precompute U1,U2 (8x8 complex), phi tables, and M_w -> d_ws.
// Kernel 2: per 16-sample tile, X(16x16) @ M_w(16x16) via v_wmma_f32_16x16x4_f32.
// ---------------------------------------------------------------------------

typedef __attribute__((ext_vector_type(2))) float v2f;
typedef __attribute__((ext_vector_type(8))) float v8f;

__constant__ float PK_LO[5] = {0.1f, 1.0f, 10.0f, 0.05f, 0.1f};
__constant__ float PK_HI[5] = {5.0f, 20.0f, 100.0f, 2.0f, 10.0f};
__constant__ float PD_LO[4] = {0.01f, 0.1f, 1.0f, 0.05f};
__constant__ float PD_HI[4] = {1.0f, 5.0f, 50.0f, 2.0f};

// ws layout (floats): [0 .. 2303] = M_w (9 matrices, 16x16 row-major)
//                     [2304..2367] = U1 real (8x8 row-major)
//                     [2368..2431] = U1 imag

__global__ __launch_bounds__(128) void pkqnc_precompute(const float* __restrict__ theta,
                                                        float* __restrict__ ws)
{
    __shared__ float U1r[64], U1i[64], U2r[64], U2i[64];
    __shared__ float ph1[9][8], ph2[9][8];

    const int tid = threadIdx.x;

    if (tid == 0) {
        float Er[3][2][2], Ei[3][2][2];
        float Ukr[64], Uki[64];
        for (int l = 0; l < 2; ++l) {
            // per-qubit combined single-qubit unitary E = RZ(e2)*RY(e1)*RX(e0)
            for (int q = 0; q < 3; ++q) {
                float e0 = theta[l*9 + q*3 + 0];
                float e1 = theta[l*9 + q*3 + 1];
                float e2 = theta[l*9 + q*3 + 2];
                float cx = cosf(0.5f*e0), sx = sinf(0.5f*e0);
                float cy = cosf(0.5f*e1), sy = sinf(0.5f*e1);
                float cz = cosf(0.5f*e2), sz = sinf(0.5f*e2);
                // T = RY * RX
                float Tr00 =  cy*cx, Ti00 =  sy*sx;
                float Tr01 = -sy*cx, Ti01 = -cy*sx;
                float Tr10 =  sy*cx, Ti10 = -cy*sx;
                float Tr11 =  cy*cx, Ti11 = -sy*sx;
                // E = RZ * T : row0 *= (cz - i sz), row1 *= (cz + i sz)
                Er[q][0][0] = Tr00*cz + Ti00*sz;  Ei[q][0][0] = Ti00*cz - Tr00*sz;
                Er[q][0][1] = Tr01*cz + Ti01*sz;  Ei[q][0][1] = Ti01*cz - Tr01*sz;
                Er[q][1][0] = Tr10*cz - Ti10*sz;  Ei[q][1][0] = Ti10*cz + Tr10*sz;
                Er[q][1][1] = Tr11*cz - Ti11*sz;  Ei[q][1][1] = Ti11*cz + Tr11*sz;
            }
            // U_kron = E0 (x) E1 (x) E2 ; index z = 4*q0 + 2*q1 + q2 (wire0 = MSB)
            for (int zp = 0; zp < 8; ++zp)
                for (int z = 0; z < 8; ++z) {
                    int a2=(zp>>2)&1, a1=(zp>>1)&1, a0=zp&1;
                    int b2=(z >>2)&1, b1=(z >>1)&1, b0=z &1;
                    float xr = Er[0][a2][b2], xi = Ei[0][a2][b2];
                    float yr = Er[1][a1][b1], yi = Ei[1][a1][b1];
                    float tr = xr*yr - xi*yi, ti = xr*yi + xi*yr;
                    float zr = Er[2][a0][b0], zi = Ei[2][a0][b0];
                    Ukr[zp*8+z] = tr*zr - ti*zi;
                    Uki[zp*8+z] = tr*zi + ti*zr;
                }
            // Apply CNOTs CX(0->1), CX(1->2), CX(2->0) as row permutations:
            // U[zp][z] = Ukron[ sig_a(sig_b(sig_c(zp))) ][z]
            for (int zp = 0; zp < 8; ++zp) {
                int p = zp ^ ((zp & 1) << 2);        // sigma_c : CX(2->0)
                p = p ^ ((p >> 1) & 1);              // sigma_b : CX(1->2)
                p = p ^ (((p >> 2) & 1) << 1);       // sigma_a : CX(0->1)
                for (int z = 0; z < 8; ++z) {
                    if (l == 0) { U1r[zp*8+z] = Ukr[p*8+z]; U1i[zp*8+z] = Uki[p*8+z]; }
                    else        { U2r[zp*8+z] = Ukr[p*8+z]; U2i[zp*8+z] = Uki[p*8+z]; }
                }
            }
        }
    }

    // Accumulated controlled-RY angles per layer/target-wire/init-basis-state.
    // theta offsets: euler 0, pk_coup 18, pd_coup 48, pk_local 72, pd_local 82.
    for (int idx = tid; idx < 144; idx += 128) {
        int l = idx / 72;
        int r = idx % 72;
        int w = r / 8;      // 0..4 = PK wires, 5..8 = PD wires
        int z = r % 8;
        float b0 = (float)((z >> 2) & 1);   // INIT_W[0] bit (MSB)
        float b1 = (float)((z >> 1) & 1);   // INIT_W[1]
        float b2 = (float)( z       & 1);   // INIT_W[2]
        float phi;
        if (w < 5) {
            phi = b0*theta[18 + l*15 + 0*5 + w]
                + b1*theta[18 + l*15 + 1*5 + w]
                + b2*theta[18 + l*15 + 2*5 + w]
                + theta[72 + l*5 + w];
        } else {
            int k = w - 5;
            phi = b0*theta[48 + l*12 + 0*4 + k]
                + b1*theta[48 + l*12 + 1*4 + k]
                + b2*theta[48 + l*12 + 2*4 + k]
                + theta[82 + l*4 + k];
        }
        if (l == 0) ph1[w][z] = phi; else ph2[w][z] = phi;
    }

    __syncthreads();

    // K_w[z1,z2] = P_w(z1,z2) * sum_zp U2[zp,z1]*conj(U2[zp,z2]) *
    //              cos(phi2_w(zp) + (phi1_w(z1)+phi1_w(z2))/2)
    // M_w = [[Kr, Ki],[-Ki, Kr]]  (16x16 real)
    for (int idx = tid; idx < 576; idx += 128) {
        int w  = idx >> 6;
        int z1 = (idx >> 3) & 7;
        int z2 = idx & 7;
        float P = 1.0f;
        for (int u = 0; u < 9; ++u)
            if (u != w) P *= cosf(0.5f * (ph1[u][z1] - ph1[u][z2]));
        float mbar = 0.5f * (ph1[w][z1] + ph1[w][z2]);
        float kr = 0.0f, ki = 0.0f;
        for (int zp = 0; zp < 8; ++zp) {
            float ur1 = U2r[zp*8+z1], ui1 = U2i[zp*8+z1];
            float ur2 = U2r[zp*8+z2], ui2 = U2i[zp*8+z2];
            float cr = ur1*ur2 + ui1*ui2;           // Re(U2[zp,z1]*conj(U2[zp,z2]))
            float ci = ui1*ur2 - ur1*ui2;           // Im(...)
            float cw = cosf(ph2[w][zp] + mbar);
            kr += cr * cw;
            ki += ci * cw;
        }
        kr *= P; ki *= P;
        float* M = ws + w*256;
        M[z1*16 + z2]         = kr;
        M[(z1+8)*16 + (z2+8)] = kr;
        M[z1*16 + (z2+8)]     = ki;
        M[(z1+8)*16 + z2]     = -ki;
    }

    // Export U1 for the per-sample encoding kernel.
    for (int i = tid; i < 64; i += 128) {
        ws[2304 + i] = U1r[i];
        ws[2368 + i] = U1i[i];
    }
}

// One wave per 16-sample tile. WMMA: D(16x16) = X(16x16) @ M_w via 4 chained
// v_wmma_f32_16x16x4_f32, then row-dot with X and bounds mapping.
__global__ __launch_bounds__(32) void pkqnc_qform(const float* __restrict__ cov,
                                                  const float* __restrict__ dose,
                                                  const float* __restrict__ ws,
                                                  float* __restrict__ out, int B)
{
    __shared__ float X[16][16];
    __shared__ float D[16][16];
    __shared__ float U1s[128];

    const int L = threadIdx.x;
    for (int i = L; i < 128; i += 32) U1s[i] = ws[2304 + i];
    __syncthreads();

    if (L < 16) {
        int b = blockIdx.x * 16 + L;
        float t0 = 0.f, t1 = 0.f, t2 = 0.f;
        if (b < B) { t0 = cov[2*b]; t1 = cov[2*b+1]; t2 = dose[b]; }
        float c0 = cosf(0.5f*t0), s0 = sinf(0.5f*t0);
        float c1 = cosf(0.5f*t1), s1 = sinf(0.5f*t1);
        float c2 = cosf(0.5f*t2), s2 = sinf(0.5f*t2);
        float p[8];
        for (int z = 0; z < 8; ++z)
            p[z] = ((z&4)?s0:c0) * ((z&2)?s1:c1) * ((z&1)?s2:c2);
        for (int zp = 0; zp < 8; ++zp) {
            float ar = 0.f, ai = 0.f;
            for (int z = 0; z < 8; ++z) {
                ar += U1s[zp*8+z]      * p[z];
                ai += U1s[64+zp*8+z]   * p[z];
            }
            X[L][zp]   = ar;   // x = [Re a ; Im a]
            X[L][8+zp] = ai;
        }
    }
    __syncthreads();

    const int half = L >> 4;     // 0: lanes 0-15, 1: lanes 16-31
    const int lr   = L & 15;

    // A-matrix (16x4 f32) fragments per ISA layout:
    // lanes 0-15: VGPR0=K0,VGPR1=K1 ; lanes 16-31: VGPR0=K2,VGPR1=K3
    v2f afrag[4];
    for (int kc = 0; kc < 4; ++kc) {
        int col = kc*4 + half*2;
        afrag[kc].x = X[lr][col];
        afrag[kc].y = X[lr][col+1];
    }

    for (int w = 0; w < 9; ++w) {
        const float* Mw = ws + w*256;
        v8f acc = {};
        for (int kc = 0; kc < 4; ++kc) {
            int krow = kc*4 + half*2;
            v2f bfrag;                         // B 4x16: dual layout of A
            bfrag.x = Mw[krow*16 + lr];
            bfrag.y = Mw[(krow+1)*16 + lr];
            acc = __builtin_amdgcn_wmma_f32_16x16x4_f32(
                false, afrag[kc], false, bfrag, (short)0, acc, false, false);
        }
        // C/D layout: lanes 0-15 hold rows 0-7, lanes 16-31 rows 8-15.
        for (int r = 0; r < 8; ++r)
            D[half*8 + r][lr] = acc[r];
        __syncthreads();
        if (L < 16) {
            int bb = blockIdx.x*16 + L;
            if (bb < B) {
                float o = 0.f;
                for (int n = 0; n < 16; ++n) o += X[L][n] * D[L][n];
                float s = 0.5f * (o + 1.0f);
                if (w < 5) out[bb*5 + w]            = PK_LO[w]   + s * (PK_HI[w]   - PK_LO[w]);
                else       out[B*5 + bb*4 + (w-5)]  = PD_LO[w-5] + s * (PD_HI[w-5] - PD_LO[w-5]);
            }
        }
        __syncthreads();
    }
}

extern "C" void kernel_launch(void* const* d_in, const int* in_sizes, int n_in,
                              void* d_out, int out_size, void* d_ws, size_t ws_size,
                              hipStream_t stream) {
    // d_in: [0]=subject_ids (unused), [1]=covariates (B,2) f32,
    //       [2]=dose_intensities (B,) f32, [3]=theta (90,) f32
    const float* cov   = (const float*)d_in[1];
    const float* dose  = (const float*)d_in[2];
    const float* theta = (const float*)d_in[3];
    float* out = (float*)d_out;
    float* ws  = (float*)d_ws;
    const int B = in_sizes[2];

    hipLaunchKernelGGL(pkqnc_precompute, dim3(1), dim3(128), 0, stream, theta, ws);
    hipLaunchKernelGGL(pkqnc_qform, dim3((B + 15) / 16), dim3(32), 0, stream,
                       cov, dose, ws, out, B);
}